// Attn_25417616458107
// MI455X (gfx1250) — compile-verified
//
#include <hip/hip_runtime.h>
#include <hip/hip_bf16.h>

// Problem constants from the reference
#define S_LEN 2048
#define B_SZ  64
#define H_DIM 1024

typedef __attribute__((ext_vector_type(2))) float v2f;
typedef __attribute__((ext_vector_type(8))) float v8f;

// ---------------------------------------------------------------------------
// Kernel 1: u[b,h] = sum_k hidden[b,k] * W[k,h]    (M=64, N=1024, K=1024)
// One wave (32 threads) computes one 16x16 output tile via V_WMMA_F32_16X16X4_F32.
// A-frag (16x4, MxK): lane<16 -> m=lane, k={k0,k0+1}; lane>=16 -> m=lane-16, k={k0+2,k0+3}
// B-frag (4x16, KxN): lane<16 -> n=lane, k={k0,k0+1}; lane>=16 -> n=lane-16, k={k0+2,k0+3}
// C/D  (16x16 f32): vgpr r -> row (r + 8*half), col = lane&15
// ---------------------------------------------------------------------------
__global__ void __launch_bounds__(32)
attn_u_wmma_kernel(const float* __restrict__ hidden,   // [B,H]
                   const float* __restrict__ W,        // [H,H] (k-major rows)
                   float* __restrict__ u)              // [B,H]
{
    const int lane  = threadIdx.x;        // 0..31 (wave32)
    const int half  = lane >> 4;          // 0 or 1
    const int l16   = lane & 15;
    const int tileN = blockIdx.x * 16;    // column tile (h)
    const int tileM = blockIdx.y * 16;    // row tile (b)

    const float* arow = hidden + (size_t)(tileM + l16) * H_DIM;

    v8f acc = {};
    for (int k0 = 0; k0 < H_DIM; k0 += 4) {
        const int ka = k0 + half * 2;
        v2f a, b;
        a[0] = arow[ka + 0];
        a[1] = arow[ka + 1];
        b[0] = W[(size_t)(ka + 0) * H_DIM + tileN + l16];
        b[1] = W[(size_t)(ka + 1) * H_DIM + tileN + l16];
        // D = A*B + C  (f32, 16x16x4)
        acc = __builtin_amdgcn_wmma_f32_16x16x4_f32(
            /*neg_a=*/false, a, /*neg_b=*/false, b,
            /*c_mod=*/(short)0, acc, /*reuse_a=*/false, /*reuse_b=*/false);
    }

#pragma unroll
    for (int r = 0; r < 8; ++r) {
        const int m = tileM + r + half * 8;
        u[(size_t)m * H_DIM + tileN + l16] = acc[r];
    }
}

// ---------------------------------------------------------------------------
// Kernel 2: scores[b,s] = sum_h u[b,h] * enc[s,b,h]
// grid = (B, S/8), block = 256 (8 waves). u[b,:] staged in LDS once per block.
// Each wave handles one s: float4 coalesced streaming of enc + wave reduction.
// ---------------------------------------------------------------------------
__global__ void __launch_bounds__(256)
attn_scores_kernel(const float* __restrict__ enc,      // [S,B,H]
                   const float* __restrict__ u,        // [B,H]
                   float* __restrict__ scores)         // [B,S]
{
    __shared__ __align__(16) float su[H_DIM];

    const int b   = blockIdx.x;
    const int tid = threadIdx.x;

    for (int i = tid; i < H_DIM; i += 256)
        su[i] = u[(size_t)b * H_DIM + i];
    __syncthreads();

    const int wave = tid >> 5;
    const int lane = tid & 31;
    const int s    = blockIdx.y * 8 + wave;

    const float4* row = (const float4*)(enc + ((size_t)s * B_SZ + b) * H_DIM);
    const float4* uu  = (const float4*)su;

    float sum = 0.0f;
#pragma unroll
    for (int i = lane; i < H_DIM / 4; i += 32) {
        // Keep the 23.3 TB/s stream fed: hint the next 512B chunk for this wave.
        __builtin_prefetch(row + i + 32, 0, 0);
        float4 v = row[i];
        float4 w = uu[i];
        sum += v.x * w.x + v.y * w.y + v.z * w.z + v.w * w.w;
    }

    // wave32 reduction
#pragma unroll
    for (int off = 16; off > 0; off >>= 1)
        sum += __shfl_xor(sum, off, 32);

    if (lane == 0)
        scores[(size_t)b * S_LEN + s] = sum;
}

// ---------------------------------------------------------------------------
// Kernel 3: out[b,0,s] = softmax_s(scores[b,s]).  One block per b.
// (The hidden.b_attn constant is uniform over s and cancels in softmax.)
// ---------------------------------------------------------------------------
__global__ void __launch_bounds__(256)
attn_softmax_kernel(const float* __restrict__ scores,  // [B,S]
                    float* __restrict__ out)           // [B,1,S]
{
    __shared__ float red[8];
    const int b   = blockIdx.x;
    const int tid = threadIdx.x;
    const int wave = tid >> 5;
    const int lane = tid & 31;
    const float* row = scores + (size_t)b * S_LEN;

    // --- max reduction ---
    float m = -3.402823466e+38f;
    for (int i = tid; i < S_LEN; i += 256)
        m = fmaxf(m, row[i]);
#pragma unroll
    for (int off = 16; off > 0; off >>= 1)
        m = fmaxf(m, __shfl_xor(m, off, 32));
    if (lane == 0) red[wave] = m;
    __syncthreads();
    if (tid == 0) {
        float mm = red[0];
#pragma unroll
        for (int w = 1; w < 8; ++w) mm = fmaxf(mm, red[w]);
        red[0] = mm;
    }
    __syncthreads();
    const float rowmax = red[0];
    __syncthreads();

    // --- sum of exp ---
    float sum = 0.0f;
    for (int i = tid; i < S_LEN; i += 256)
        sum += __expf(row[i] - rowmax);
#pragma unroll
    for (int off = 16; off > 0; off >>= 1)
        sum += __shfl_xor(sum, off, 32);
    if (lane == 0) red[wave] = sum;
    __syncthreads();
    if (tid == 0) {
        float ss = 0.0f;
#pragma unroll
        for (int w = 0; w < 8; ++w) ss += red[w];
        red[0] = ss;
    }
    __syncthreads();
    const float inv = 1.0f / red[0];

    for (int i = tid; i < S_LEN; i += 256)
        out[(size_t)b * S_LEN + i] = __expf(row[i] - rowmax) * inv;
}

// ---------------------------------------------------------------------------
extern "C" void kernel_launch(void* const* d_in, const int* in_sizes, int n_in,
                              void* d_out, int out_size, void* d_ws, size_t ws_size,
                              hipStream_t stream) {
    const float* hidden = (const float*)d_in[0];   // [1,B,H]
    const float* enc    = (const float*)d_in[1];   // [S,B,H]
    const float* W      = (const float*)d_in[2];   // [H,H]
    // d_in[3] = b_attn: uniform-over-s contribution, cancels in softmax.

    float* u      = (float*)d_ws;                          // B*H floats (256 KB)
    float* scores = (float*)d_ws + (size_t)B_SZ * H_DIM;   // B*S floats (512 KB)
    float* out    = (float*)d_out;                         // [B,1,S]

    // 1) u = hidden @ W  via f32 WMMA (16x16 tiles, K in steps of 4)
    attn_u_wmma_kernel<<<dim3(H_DIM / 16, B_SZ / 16), dim3(32), 0, stream>>>(hidden, W, u);

    // 2) scores[b,s] = <u[b,:], enc[s,b,:]>  — the 512 MB HBM-bound stream
    attn_scores_kernel<<<dim3(B_SZ, S_LEN / 8), dim3(256), 0, stream>>>(enc, u, scores);

    // 3) row softmax
    attn_softmax_kernel<<<dim3(B_SZ), dim3(256), 0, stream>>>(scores, out);
}